// RecurrentCycle_49615462203474
// MI455X (gfx1250) — compile-verified
//
#include <hip/hip_runtime.h>
#include <stdint.h>

// Problem constants (fixed by the reference).
#ifndef CYCLE_LEN
#define CYCLE_LEN 168
#endif
#define CHANNELS   64
#define OUT_LEN    720

typedef __attribute__((ext_vector_type(4))) float v4f;

// 168 rows x 64 floats = 2688 float4 = 43008 bytes of LDS.
#define LUT_FLOATS   (CYCLE_LEN * CHANNELS)
#define LUT_VEC4     (LUT_FLOATS / 4)          // 2688
#define STAGE_ROUNDS ((LUT_VEC4 + 255) / 256)  // 11

__global__ __launch_bounds__(256) void
RecurrentCycle_gather_kernel(const int* __restrict__ cycle_index,
                             const float* __restrict__ data,
                             float* __restrict__ out)
{
    __shared__ alignas(16) float lut[LUT_FLOATS];

    const int tid = threadIdx.x;
    const int b   = blockIdx.x;

    // ---- Stage the 43KB table into LDS with CDNA5 async global->LDS copies.
    {
        const uint64_t gbase = (uint64_t)(const void*)data;
        const unsigned lbase = (unsigned)(uintptr_t)(&lut[0]); // LDS byte offset
#pragma unroll
        for (int i = 0; i < STAGE_ROUNDS; ++i) {
            const int g = i * 256 + tid;            // float4 index
            if (g < LUT_VEC4) {
                const unsigned lds_off = lbase + (unsigned)(g * 16);
                const uint64_t gaddr   = gbase + (uint64_t)g * 16u;
                // ASYNCcnt-tracked DMA-style copy: 16B per lane -> LDS.
                asm volatile("global_load_async_to_lds_b128 %0, %1, off"
                             :: "v"(lds_off), "v"(gaddr)
                             : "memory");
            }
        }
    }
    // Drain ASYNCcnt, then make the staged table visible to the workgroup.
    asm volatile("s_wait_asynccnt 0x0" ::: "memory");
    __syncthreads();

    // ---- Per-batch start phase (uniform per block -> scalarized load).
    int s = cycle_index[b] % CYCLE_LEN;
    if (s < 0) s += CYCLE_LEN;

    // Thread mapping: 16 threads cover one 64-float output row (4 floats each).
    const int rg = tid >> 4;   // row within group of 16 rows
    const int cg = tid & 15;   // 4-channel group

    int r = s + rg;
    if (r >= CYCLE_LEN) r -= CYCLE_LEN;

    float* __restrict__ optr =
        out + (size_t)b * (OUT_LEN * CHANNELS) + (size_t)rg * CHANNELS + cg * 4;

    const float* lsrc_base = &lut[cg * 4];

#pragma unroll 5
    for (int l0 = 0; l0 < OUT_LEN; l0 += 16) {
        // ds_load_b128 (16B aligned: row stride = 64 dwords)
        const v4f v = *(const v4f*)(lsrc_base + r * CHANNELS);
        // Streaming 755MB output: non-temporal b128 store, fully coalesced
        // (a wave32 writes 512 contiguous bytes per iteration).
        __builtin_nontemporal_store(v, (v4f*)optr);

        optr += 16 * CHANNELS;
        r += 16;
        if (r >= CYCLE_LEN) r -= CYCLE_LEN;
    }
}

extern "C" void kernel_launch(void* const* d_in, const int* in_sizes, int n_in,
                              void* d_out, int out_size, void* d_ws, size_t ws_size,
                              hipStream_t stream)
{
    // setup_inputs() order: cycle_index (int32[B]), output_len (int32[1]), data (f32[168*64])
    const int*   cyc  = (const int*)d_in[0];
    const float* data = (const float*)d_in[2];
    float*       out  = (float*)d_out;

    const int B = in_sizes[0]; // 4096

    RecurrentCycle_gather_kernel<<<B, 256, 0, stream>>>(cyc, data, out);
}